// TorsionalPrior_88175678587352
// MI455X (gfx1250) — compile-verified
//
#include <hip/hip_runtime.h>
#include <math.h>

#define N_NODE        1000000
#define N_BOND        100000
#define K_TWIST       40
#define MAX_ORDER     8
#define N_GROUP       (N_BOND / MAX_ORDER)      // 12500
#define N_DOMAIN      10000
#define ROWS_PER_DOM  (N_NODE / N_DOMAIN)       // 100
#define TILES_PER_DOM 7                         // ceil(100/16)
#define SIGMA_MAX_F   3.14159265358979323846f

typedef __attribute__((ext_vector_type(2))) float v2f;
typedef __attribute__((ext_vector_type(8))) float v8f;

// ---------------------------------------------------------------------------
// Phase A (per order): snapshot Rodrigues matrix + pivot for the active bond
// of each group into rodWS[g*12 + 0..11] = {R row-major, pivot}.
// Must run as its own kernel: axis endpoints may be nodes rewritten in the
// same order's apply phase, so the snapshot needs a global sync before apply.
// ---------------------------------------------------------------------------
__global__ void tor_build_R(const float* __restrict__ posTor,
                            const float* __restrict__ info_level,
                            const int*   __restrict__ tor_bonds,
                            const float* __restrict__ eps,
                            const float* __restrict__ uniform_angles,
                            const int*   __restrict__ from_prior,
                            float* __restrict__ rodWS,
                            int order)
{
    int g = blockIdx.x * blockDim.x + threadIdx.x;
    if (g >= N_GROUP) return;
    int b = g * MAX_ORDER + order;          // active bond: order == b % 8
    int u = tor_bonds[b * 3 + 1];
    int v = tor_bonds[b * 3 + 2];
    float ux = posTor[u*3+0], uy = posTor[u*3+1], uz = posTor[u*3+2];
    float vx = posTor[v*3+0], vy = posTor[v*3+1], vz = posTor[v*3+2];
    float ax = vx - ux, ay = vy - uy, az = vz - uz;
    float inv = 1.0f / (sqrtf(ax*ax + ay*ay + az*az) + 1e-12f);
    ax *= inv; ay *= inv; az *= inv;

    float info = info_level[b];
    float ang  = eps[b] * (1.0f - info) * SIGMA_MAX_F;
    if (from_prior[0] != 0 && info == 0.0f) ang = uniform_angles[b];

    float s, c;
    sincosf(ang, &s, &c);
    float t = 1.0f - c;

    float* W = rodWS + g * 12;
    W[0] = c + t*ax*ax;      W[1] = t*ax*ay - s*az;   W[2] = t*ax*az + s*ay;
    W[3] = t*ax*ay + s*az;   W[4] = c + t*ay*ay;      W[5] = t*ay*az - s*ax;
    W[6] = t*ax*az - s*ay;   W[7] = t*ay*az + s*ax;   W[8] = c + t*az*az;
    W[9] = vx; W[10] = vy; W[11] = vz;                // pivot = pos_tor[v]
}

// ---------------------------------------------------------------------------
// Phase B (per order): rotate the 40 twisted nodes of every active bond about
// its snapshotted pivot. Node indices are unique within an order -> no races.
// ---------------------------------------------------------------------------
__global__ void tor_apply(float* __restrict__ posTor,
                          const int* __restrict__ twisted,   // (N_BOND*K_TWIST, 2)
                          const float* __restrict__ rodWS,
                          int order)
{
    int t = blockIdx.x * blockDim.x + threadIdx.x;
    if (t >= N_GROUP * K_TWIST) return;
    int g = t / K_TWIST;
    int k = t - g * K_TWIST;
    int b = g * MAX_ORDER + order;
    int e = b * K_TWIST + k;                 // entry in twisted_nodes_anno
    int node = twisted[e * 2 + 1];

    const float* W = rodWS + g * 12;
    float px = posTor[node*3+0] - W[9];
    float py = posTor[node*3+1] - W[10];
    float pz = posTor[node*3+2] - W[11];
    float rx = W[0]*px + W[1]*py + W[2]*pz + W[9];
    float ry = W[3]*px + W[4]*py + W[5]*pz + W[10];
    float rz = W[6]*px + W[7]*py + W[8]*pz + W[11];
    posTor[node*3+0] = rx;
    posTor[node*3+1] = ry;
    posTor[node*3+2] = rz;
}

// ---------------------------------------------------------------------------
// Kabsch per domain: one wave32 per domain. Reduce 16 accumulators via
// __shfl_down, then lane 0 does Jacobi eigensolve of H^T H and builds
// R = v1 u1^T + v2 u2^T + v3 u3^T (right-handed frames -> reflection fix
// folds in automatically, identical to reference's det-sign correction).
// kabWS[dom*12 + 0..11] = {R row-major, t}.
// ---------------------------------------------------------------------------
__global__ void kabsch_domains(const float* __restrict__ posTor,
                               const float* __restrict__ pos,
                               const int* __restrict__ domain_index,
                               const int* __restrict__ node_index,
                               float* __restrict__ kabWS)
{
    int wave = (blockIdx.x * blockDim.x + threadIdx.x) >> 5;
    int lane = threadIdx.x & 31;
    if (wave >= N_DOMAIN) return;
    int base = wave * ROWS_PER_DOM;

    float acc[16];
#pragma unroll
    for (int i = 0; i < 16; ++i) acc[i] = 0.0f;

    for (int i = base + lane; i < base + ROWS_PER_DOM; i += 32) {
        int n = node_index[i];
        float Px = posTor[n*3+0], Py = posTor[n*3+1], Pz = posTor[n*3+2];
        float Qx = pos[n*3+0],    Qy = pos[n*3+1],    Qz = pos[n*3+2];
        acc[0] += 1.0f;
        acc[1] += Px;  acc[2] += Py;  acc[3] += Pz;
        acc[4] += Qx;  acc[5] += Qy;  acc[6] += Qz;
        acc[7]  += Px*Qx; acc[8]  += Px*Qy; acc[9]  += Px*Qz;
        acc[10] += Py*Qx; acc[11] += Py*Qy; acc[12] += Py*Qz;
        acc[13] += Pz*Qx; acc[14] += Pz*Qy; acc[15] += Pz*Qz;
    }
#pragma unroll
    for (int off = 16; off > 0; off >>= 1) {
#pragma unroll
        for (int k = 0; k < 16; ++k)
            acc[k] += __shfl_down(acc[k], off);
    }
    if (lane != 0) return;

    float invc = 1.0f / acc[0];
    float sP[3] = {acc[1], acc[2], acc[3]};
    float sQ[3] = {acc[4], acc[5], acc[6]};
    float cP[3] = {sP[0]*invc, sP[1]*invc, sP[2]*invc};
    float cQ[3] = {sQ[0]*invc, sQ[1]*invc, sQ[2]*invc};
    float H[3][3];
#pragma unroll
    for (int r = 0; r < 3; ++r)
#pragma unroll
        for (int c = 0; c < 3; ++c)
            H[r][c] = acc[7 + r*3 + c] - sP[r]*sQ[c]*invc;

    // A = H^T H (symmetric PSD), cyclic Jacobi, accumulate V
    float A[3][3], V[3][3] = {{1,0,0},{0,1,0},{0,0,1}};
#pragma unroll
    for (int i = 0; i < 3; ++i)
#pragma unroll
        for (int j = 0; j < 3; ++j)
            A[i][j] = H[0][i]*H[0][j] + H[1][i]*H[1][j] + H[2][i]*H[2][j];

    const int PP[3] = {0, 0, 1};
    const int QQ[3] = {1, 2, 2};
    for (int sweep = 0; sweep < 6; ++sweep) {
        for (int r = 0; r < 3; ++r) {
            int p = PP[r], q = QQ[r];
            float apq = A[p][q];
            if (fabsf(apq) > 1e-18f) {
                float tau = (A[q][q] - A[p][p]) / (2.0f * apq);
                float tt  = copysignf(1.0f, tau) / (fabsf(tau) + sqrtf(1.0f + tau*tau));
                float cj  = 1.0f / sqrtf(1.0f + tt*tt);
                float sj  = tt * cj;
                for (int k = 0; k < 3; ++k) {
                    float akp = A[k][p], akq = A[k][q];
                    A[k][p] = cj*akp - sj*akq;
                    A[k][q] = sj*akp + cj*akq;
                }
                for (int k = 0; k < 3; ++k) {
                    float apk = A[p][k], aqk = A[q][k];
                    A[p][k] = cj*apk - sj*aqk;
                    A[q][k] = sj*apk + cj*aqk;
                }
                for (int k = 0; k < 3; ++k) {
                    float vkp = V[k][p], vkq = V[k][q];
                    V[k][p] = cj*vkp - sj*vkq;
                    V[k][q] = sj*vkp + cj*vkq;
                }
            }
        }
    }
    // sort eigenpairs descending (network on 3)
    float lam[3] = {A[0][0], A[1][1], A[2][2]};
    auto cswap = [&](int a, int b) {
        if (lam[a] < lam[b]) {
            float tl = lam[a]; lam[a] = lam[b]; lam[b] = tl;
            for (int k = 0; k < 3; ++k) { float tv = V[k][a]; V[k][a] = V[k][b]; V[k][b] = tv; }
        }
    };
    cswap(0, 1); cswap(0, 2); cswap(1, 2);

    float e1[3] = {V[0][0], V[1][0], V[2][0]};
    float e2[3] = {V[0][1], V[1][1], V[2][1]};
    float e3[3] = {e1[1]*e2[2] - e1[2]*e2[1],
                   e1[2]*e2[0] - e1[0]*e2[2],
                   e1[0]*e2[1] - e1[1]*e2[0]};            // right-handed V

    float u1[3], u2[3], u3[3];
    for (int i = 0; i < 3; ++i) u1[i] = H[i][0]*e1[0] + H[i][1]*e1[1] + H[i][2]*e1[2];
    float n1 = sqrtf(u1[0]*u1[0] + u1[1]*u1[1] + u1[2]*u1[2]);
    if (n1 > 1e-12f) { float s = 1.0f/n1; u1[0]*=s; u1[1]*=s; u1[2]*=s; }
    else { u1[0] = e1[0]; u1[1] = e1[1]; u1[2] = e1[2]; }
    for (int i = 0; i < 3; ++i) u2[i] = H[i][0]*e2[0] + H[i][1]*e2[1] + H[i][2]*e2[2];
    float d12 = u2[0]*u1[0] + u2[1]*u1[1] + u2[2]*u1[2];
    for (int i = 0; i < 3; ++i) u2[i] -= d12 * u1[i];
    float n2 = sqrtf(u2[0]*u2[0] + u2[1]*u2[1] + u2[2]*u2[2]);
    if (n2 > 1e-12f) { float s = 1.0f/n2; u2[0]*=s; u2[1]*=s; u2[2]*=s; }
    else {
        float ex[3] = { (fabsf(u1[0]) < 0.9f) ? 1.0f : 0.0f,
                        (fabsf(u1[0]) < 0.9f) ? 0.0f : 1.0f, 0.0f };
        u2[0] = u1[1]*ex[2] - u1[2]*ex[1];
        u2[1] = u1[2]*ex[0] - u1[0]*ex[2];
        u2[2] = u1[0]*ex[1] - u1[1]*ex[0];
        float nn = 1.0f / sqrtf(u2[0]*u2[0] + u2[1]*u2[1] + u2[2]*u2[2] + 1e-30f);
        u2[0]*=nn; u2[1]*=nn; u2[2]*=nn;
    }
    u3[0] = u1[1]*u2[2] - u1[2]*u2[1];
    u3[1] = u1[2]*u2[0] - u1[0]*u2[2];
    u3[2] = u1[0]*u2[1] - u1[1]*u2[0];

    float R[3][3];
    for (int i = 0; i < 3; ++i)
        for (int j = 0; j < 3; ++j)
            R[i][j] = e1[i]*u1[j] + e2[i]*u2[j] + e3[i]*u3[j];

    int dom = domain_index[base];
    float* W = kabWS + dom * 12;
    for (int i = 0; i < 3; ++i)
        for (int j = 0; j < 3; ++j)
            W[i*3 + j] = R[i][j];
    for (int i = 0; i < 3; ++i)
        W[9 + i] = cQ[i] - (R[i][0]*cP[0] + R[i][1]*cP[1] + R[i][2]*cP[2]);
}

// ---------------------------------------------------------------------------
// Final affine apply via V_WMMA_F32_16X16X4_F32: D = A(16x4) x B(4x16).
// One wave per 16-row tile of a domain. A rows = [x y z 1] of 16 nodes
// (padded rows zeroed); B cols 0..2 = [R^T ; t^T] so D[m][j] = (R P_m + t)[j].
// A layout (ISA 7.12.2, 32-bit 16x4): lane<16 holds row=lane with K0(v0)/K1(v1);
// lane>=16 holds row=lane-16 with K2(v0)/K3(v1). B mirrors (K split by lane half).
// In-place on posTor is safe: tiles partition the nodes; loads feed the WMMA,
// so they complete before the dependent stores.
// ---------------------------------------------------------------------------
__global__ void apply_rigid_wmma(float* __restrict__ posTor,
                                 const int* __restrict__ domain_index,
                                 const int* __restrict__ node_index,
                                 const float* __restrict__ kabWS)
{
    int wave = (blockIdx.x * blockDim.x + threadIdx.x) >> 5;
    int lane = threadIdx.x & 31;
    if (wave >= N_DOMAIN * TILES_PER_DOM) return;   // wave-uniform

    int domSlot = wave / TILES_PER_DOM;
    int tile    = wave - domSlot * TILES_PER_DOM;
    int base    = domSlot * ROWS_PER_DOM + tile * 16;
    int nrows   = ROWS_PER_DOM - tile * 16;
    if (nrows > 16) nrows = 16;

    int dom = domain_index[base];
    const float* M = kabWS + dom * 12;              // R row-major + t

    bool hi = (lane >= 16);
    int  r  = lane & 15;
    int  rc = (r < nrows) ? r : 0;
    int  na = node_index[base + rc];
    float x = posTor[na*3+0], y = posTor[na*3+1], z = posTor[na*3+2];
    bool rv = (r < nrows);
    v2f a;
    a.x = rv ? (hi ? z : x)    : 0.0f;              // K2 : K0
    a.y = rv ? (hi ? 1.0f : y) : 0.0f;              // K3 : K1

    int  j  = lane & 15;                            // output column (coordinate)
    int  jc = (j < 3) ? j : 0;
    float m0 = M[jc*3 + 0];                         // B[0][j] = R[j][0]
    float m1 = M[jc*3 + 1];                         // B[1][j] = R[j][1]
    float m2 = M[jc*3 + 2];                         // B[2][j] = R[j][2]
    float m3 = M[9 + jc];                           // B[3][j] = t[j]
    bool jv = (j < 3);
    v2f b;
    b.x = jv ? (hi ? m2 : m0) : 0.0f;               // K2 : K0
    b.y = jv ? (hi ? m3 : m1) : 0.0f;               // K3 : K1

    v8f c = {};
    v8f d = __builtin_amdgcn_wmma_f32_16x16x4_f32(
        /*neg_a=*/false, a, /*neg_b=*/false, b,
        /*c_mod=*/(short)0, c, /*reuse_a=*/false, /*reuse_b=*/false);

    // D layout: vgpr rr, lane<16 -> row rr, lane>=16 -> row rr+8; N = lane&15
    if (jv) {
        int mbase = hi ? 8 : 0;
#pragma unroll
        for (int rr = 0; rr < 8; ++rr) {
            int m = mbase + rr;
            if (m < nrows) {
                int n = node_index[base + m];
                posTor[n*3 + j] = d[rr];
            }
        }
    }
}

// ---------------------------------------------------------------------------
extern "C" void kernel_launch(void* const* d_in, const int* in_sizes, int n_in,
                              void* d_out, int out_size, void* d_ws, size_t ws_size,
                              hipStream_t stream)
{
    (void)in_sizes; (void)n_in; (void)out_size; (void)ws_size;
    const float* pos        = (const float*)d_in[0];
    const float* info_level = (const float*)d_in[1];
    const int*   tor_bonds  = (const int*)d_in[2];
    const int*   twisted    = (const int*)d_in[3];
    const int*   dom_idx    = (const int*)d_in[4];
    const int*   node_idx   = (const int*)d_in[5];
    const float* eps        = (const float*)d_in[6];
    const float* uang       = (const float*)d_in[7];
    const int*   from_prior = (const int*)d_in[8];

    float* posTor = (float*)d_out;                  // working pos_tor buffer
    float* rodWS  = (float*)d_ws;                   // N_GROUP*12 floats
    float* kabWS  = rodWS + N_GROUP * 12;           // N_DOMAIN*12 floats

    hipMemcpyAsync(posTor, pos, sizeof(float) * 3 * N_NODE,
                   hipMemcpyDeviceToDevice, stream);

    for (int o = 0; o < MAX_ORDER; ++o) {
        tor_build_R<<<(N_GROUP + 255) / 256, 256, 0, stream>>>(
            posTor, info_level, tor_bonds, eps, uang, from_prior, rodWS, o);
        tor_apply<<<(N_GROUP * K_TWIST + 255) / 256, 256, 0, stream>>>(
            posTor, twisted, rodWS, o);
    }
    kabsch_domains<<<(N_DOMAIN * 32 + 255) / 256, 256, 0, stream>>>(
        posTor, pos, dom_idx, node_idx, kabWS);
    apply_rigid_wmma<<<(N_DOMAIN * TILES_PER_DOM * 32 + 255) / 256, 256, 0, stream>>>(
        posTor, dom_idx, node_idx, kabWS);
}